// NeuralODEClassifier_46789373723010
// MI455X (gfx1250) — compile-verified
//
#include <hip/hip_runtime.h>
#include <math.h>

// ---------------- problem constants ----------------
#define B_TOT   262144
#define HIDDEN  96
#define NWAVE   8                         // waves per block (256 threads)
#define ELEMS_PER_BLOCK (16 * NWAVE)      // 128 batch elements per block
#define NBLK    (B_TOT / ELEMS_PER_BLOCK) // 2048 blocks for bigstep
#define PS2     112                       // W2 pair-row stride in float2 units (2*PS2 % 64 == 32 banks)
#define HS      100                       // padded LDS stride for h-tile rows (conflict-free, 16B-aligned rows)
#define MAX_STEPS 60

typedef __attribute__((ext_vector_type(2))) float v2f;
typedef __attribute__((ext_vector_type(4))) float v4f;
typedef __attribute__((ext_vector_type(8))) float v8f;

__device__ __forceinline__ float silu_f(float x) {
  // x * sigmoid(x); v_exp_f32 + v_rcp_f32
  return x * __builtin_amdgcn_rcpf(1.0f + __expf(-x));
}

// ---------------- init / encode / commit / decode ----------------

__global__ void init_ws_kernel(float* ws) {
  // ws[0]=t  ws[1]=dt  ws[2]=dt_c  ws[3]=done  ws[4]=commit
  ws[0] = 0.0f;
  ws[1] = 0.05f;
  ws[2] = 0.05f;   // min(dt, T1 - t) at t=0
  ws[3] = 0.0f;
  ws[4] = 0.0f;
}

__global__ void encode_kernel(const float* __restrict__ x, float* __restrict__ y) {
  int e = blockIdx.x * blockDim.x + threadIdx.x;
  float v = x[e];
  y[e * 2 + 0] = v;
  y[e * 2 + 1] = v * v;   // 'parabola' encoder
}

__global__ void commit_kernel(float* __restrict__ y, const float* __restrict__ y5,
                              const float* __restrict__ ws) {
  if (ws[4] == 0.0f) return;
  int i = blockIdx.x * blockDim.x + threadIdx.x;
  y[i] = y5[i];
}

__global__ void decode_kernel(const float* __restrict__ y, const float* __restrict__ Wd,
                              const float* __restrict__ bd, float* __restrict__ out) {
  int e = blockIdx.x * blockDim.x + threadIdx.x;
  float y0 = y[e * 2 + 0];
  float y1 = y[e * 2 + 1];
#pragma unroll
  for (int j = 0; j < 10; ++j)
    out[e * 10 + j] = fmaf(y0, Wd[j], fmaf(y1, Wd[10 + j], bd[j]));
}

// ---------------- control: global error norm + adaptive dt ----------------

__global__ __launch_bounds__(256) void control_kernel(float* ws) {
  __shared__ float red[256];
  const int tid = threadIdx.x;
  float s = 0.0f;
  const float* part = ws + 16;
  for (int i = tid; i < NBLK; i += 256) s += part[i];   // fixed order: deterministic
  red[tid] = s;
  __syncthreads();
  for (int w = 128; w > 0; w >>= 1) {
    if (tid < w) red[tid] += red[tid + w];
    __syncthreads();
  }
  if (tid == 0) {
    if (ws[3] != 0.0f) {        // done: freeze everything, never commit again
      ws[4] = 0.0f;
    } else {
      float t   = ws[0];
      float dtc = ws[2];
      float enorm = sqrtf(red[0] / (float)(B_TOT * 2));
      int accept = (enorm <= 1.0f) ? 1 : 0;
      float factor = 0.9f * powf(fmaxf(enorm, 1e-10f), -0.2f);
      factor = fminf(fmaxf(factor, 0.2f), 10.0f);
      if (accept) t = t + dtc;
      float dt = dtc * factor;
      ws[0] = t;
      ws[1] = dt;
      ws[3] = (t >= 1.0f - 1e-7f) ? 1.0f : 0.0f;  // done for next iter
      ws[2] = fminf(dt, 1.0f - t);                // dt_c for next iter
      ws[4] = accept ? 1.0f : 0.0f;
    }
  }
}

// ---------------- one dopri5 step: 7 MLP evals, fp32 WMMA ----------------

__global__ __launch_bounds__(256) void bigstep_kernel(
    const float* __restrict__ W1, const float* __restrict__ b1,
    const float* __restrict__ W2, const float* __restrict__ b2,
    const float* __restrict__ W3, const float* __restrict__ b3,
    float* __restrict__ ws, const float* __restrict__ y,
    float* __restrict__ y5out)
{
  // Butcher tableau (dopri5), folded at compile time (stage loop fully unrolled)
  constexpr float AA[7][6] = {
    {0.f, 0.f, 0.f, 0.f, 0.f, 0.f},
    {0.2f, 0.f, 0.f, 0.f, 0.f, 0.f},
    {3.f/40.f, 9.f/40.f, 0.f, 0.f, 0.f, 0.f},
    {44.f/45.f, -56.f/15.f, 32.f/9.f, 0.f, 0.f, 0.f},
    {19372.f/6561.f, -25360.f/2187.f, 64448.f/6561.f, -212.f/729.f, 0.f, 0.f},
    {9017.f/3168.f, -355.f/33.f, 46732.f/5247.f, 49.f/176.f, -5103.f/18656.f, 0.f},
    {35.f/384.f, 0.f, 500.f/1113.f, 125.f/192.f, -2187.f/6784.f, 11.f/84.f},
  };
  constexpr float CC[7]  = {0.f, 0.2f, 0.3f, 0.8f, 8.f/9.f, 1.f, 1.f};
  constexpr float B5c[7] = {35.f/384.f, 0.f, 500.f/1113.f, 125.f/192.f, -2187.f/6784.f, 11.f/84.f, 0.f};
  constexpr float B4c[7] = {5179.f/57600.f, 0.f, 7571.f/16695.f, 393.f/640.f,
                            -92097.f/339200.f, 187.f/2100.f, 1.f/40.f};

  // W1aug pairs: {row0,row1} and {row2,row3} interleaved -> one b64 per B operand
  __shared__ __align__(16) float w1p[2 * 96 * 2];
  __shared__ float b2s[96];
  // W2 row pairs interleaved: w2p[(pair*PS2 + col)*2 + sub] = W2[2*pair+sub][col]
  __shared__ __align__(16) float w2p[48 * PS2 * 2];
  // W3 transposed: w3t[j*96 + n] = W3[n][j]  -> contiguous b128 reads along n
  __shared__ __align__(16) float w3t[2 * 96];
  __shared__ float b3s[2];
  __shared__ __align__(16) float hbuf[NWAVE][16 * HS];  // per-wave 16x96 tile (padded)
  __shared__ float red[256];

  const float done = ws[3];
  if (done != 0.0f) return;                // uniform branch, all threads
  const float t0  = ws[0];
  const float dtc = ws[2];

  const int tid = threadIdx.x;
  // --- preload weights into LDS (pair-interleaved layouts) ---
  for (int i = tid; i < 2 * 96; i += 256) {          // W1aug: rows 0..2 = W1, row 3 = b1
    int pr = i / 96, c = i % 96;                     // pr=0 -> rows(0,1), pr=1 -> rows(2,3)
    float lo = W1[(2 * pr + 0) * 96 + c];
    float hi_v = (pr == 0) ? W1[96 + c] : b1[c];     // row1=W1[1], row3=b1
    w1p[(pr * 96 + c) * 2 + 0] = lo;
    w1p[(pr * 96 + c) * 2 + 1] = hi_v;
  }
  for (int i = tid; i < 96; i += 256) b2s[i] = b2[i];
  for (int i = tid; i < 96 * 96; i += 256) {
    int r = i / 96, c = i % 96;
    w2p[((r >> 1) * PS2 + c) * 2 + (r & 1)] = W2[i];
  }
  for (int i = tid; i < 192; i += 256) {
    int j = i / 96, n = i % 96;
    w3t[j * 96 + n] = W3[n * 2 + j];
  }
  if (tid < 2) b3s[tid] = b3[tid];
  __syncthreads();

  const int wave = tid >> 5;
  const int lane = tid & 31;
  const int m    = lane & 15;   // element-in-tile (also N index within a 16-col tile)
  const int hi   = lane >> 4;   // half-wave selector (K/M offset per WMMA layout)
  const int elem = blockIdx.x * ELEMS_PER_BLOCK + wave * 16 + m;

  const float yv = y[elem * 2 + hi];       // this lane owns y[elem][hi]
  float* hb = &hbuf[wave][0];

  float ks[7];
#pragma unroll
  for (int i = 0; i < 7; ++i) {
    // stage input: yi = y + dt * sum_j A[i][j] * k_j   (per-lane scalar)
    float yi = yv;
#pragma unroll
    for (int j = 0; j < 6; ++j)
      if (j < i) yi = fmaf(dtc * AA[i][j], ks[j], yi);
    const float ti = t0 + CC[i] * dtc;

    // ---- layer 1: [y0,y1,t,1](16x4) @ [W1;b1](4x96), one WMMA per 16-col tile
    // A layout (16x4 f32): lanes 0-15: V0=K0,V1=K1 ; lanes 16-31: V0=K2,V1=K3
    float partner = __shfl_xor(yi, 16, 32);   // lanes<16 get yi of dim=1
    v2f a1;
    a1.x = hi ? ti : yi;
    a1.y = hi ? 1.0f : partner;
#pragma unroll
    for (int c = 0; c < 6; ++c) {
      const int col = c * 16 + m;
      const v2f bw = *(const v2f*)&w1p[(hi * 96 + col) * 2];  // one b64
      v8f acc = {};
      acc = __builtin_amdgcn_wmma_f32_16x16x4_f32(false, a1, false, bw,
                                                  (short)0, acc, false, false);
#pragma unroll
      for (int v = 0; v < 8; ++v)                      // D layout -> LDS row-major
        hb[(v + 8 * hi) * HS + col] = silu_f(acc[v]);
    }
    asm volatile("s_wait_dscnt 0" ::: "memory");       // wave's LDS stores visible

    // ---- layer 2: h1(16x96) @ W2(96x96) + b2 : 24 K-tiles x 6 N-tiles of WMMA
    v8f acc2[6];
#pragma unroll
    for (int c = 0; c < 6; ++c) {
      const float bb = b2s[c * 16 + m];                // bias broadcast along M
#pragma unroll
      for (int v = 0; v < 8; ++v) acc2[c][v] = bb;
    }
#pragma unroll 4
    for (int k = 0; k < 24; ++k) {
      const v2f a2 = *(const v2f*)&hb[m * HS + 4 * k + 2 * hi];  // one b64, conflict-free
      const int pidx = 2 * k + hi;                               // W2 row pair index
#pragma unroll
      for (int c = 0; c < 6; ++c) {
        const v2f bw = *(const v2f*)&w2p[(pidx * PS2 + c * 16 + m) * 2];  // one b64
        acc2[c] = __builtin_amdgcn_wmma_f32_16x16x4_f32(false, a2, false, bw,
                                                        (short)0, acc2[c], false, false);
      }
    }
    asm volatile("s_wait_dscnt 0" ::: "memory");       // all A-reads done before overwrite
#pragma unroll
    for (int c = 0; c < 6; ++c)
#pragma unroll
      for (int v = 0; v < 8; ++v)
        hb[(v + 8 * hi) * HS + c * 16 + m] = silu_f(acc2[c][v]);
    asm volatile("s_wait_dscnt 0" ::: "memory");

    // ---- layer 3: h2(16x96) @ W3(96x2) + b3 ; lane -> one (elem, dim) output
    // b128 vector reads: h2 row (16B-aligned, broadcast across half-waves) x w3 column
    float a3 = b3s[hi];
    const v4f* hrow = (const v4f*)&hb[m * HS];         // HS*4 = 400B row stride
    const v4f* wcol = (const v4f*)&w3t[hi * 96];
#pragma unroll
    for (int n4 = 0; n4 < 24; ++n4) {
      v4f hv = hrow[n4];
      v4f wv = wcol[n4];
      a3 = fmaf(hv.x, wv.x, a3);
      a3 = fmaf(hv.y, wv.y, a3);
      a3 = fmaf(hv.z, wv.z, a3);
      a3 = fmaf(hv.w, wv.w, a3);
    }
    ks[i] = a3;
    asm volatile("s_wait_dscnt 0" ::: "memory");       // reads done before next stage's stores
  }

  // ---- combine stages: 5th-order solution + embedded error
  float y5 = yv, ev = 0.0f;
#pragma unroll
  for (int i = 0; i < 7; ++i) {
    y5 = fmaf(dtc * B5c[i], ks[i], y5);
    ev = fmaf(dtc * (B5c[i] - B4c[i]), ks[i], ev);
  }
  y5out[elem * 2 + hi] = y5;

  const float sc = 1e-4f + 1e-4f * fmaxf(fabsf(yv), fabsf(y5));
  const float r = ev / sc;

  // deterministic per-block partial sum of (err/scale)^2
  red[tid] = r * r;
  __syncthreads();
  for (int w = 128; w > 0; w >>= 1) {
    if (tid < w) red[tid] += red[tid + w];
    __syncthreads();
  }
  if (tid == 0) ws[16 + blockIdx.x] = red[0];
}

// ---------------- host-side orchestration ----------------

extern "C" void kernel_launch(void* const* d_in, const int* in_sizes, int n_in,
                              void* d_out, int out_size, void* d_ws, size_t ws_size,
                              hipStream_t stream) {
  (void)in_sizes; (void)n_in; (void)out_size; (void)ws_size;
  const float* x  = (const float*)d_in[0];
  const float* W1 = (const float*)d_in[1];
  const float* b1 = (const float*)d_in[2];
  const float* W2 = (const float*)d_in[3];
  const float* b2 = (const float*)d_in[4];
  const float* W3 = (const float*)d_in[5];
  const float* b3 = (const float*)d_in[6];
  const float* Wd = (const float*)d_in[7];
  const float* bd = (const float*)d_in[8];
  float* out = (float*)d_out;
  float* ws  = (float*)d_ws;

  float* yst = ws + 16 + NBLK;   // committed state y, B x 2 floats in d_ws
  float* y5  = out;              // candidate y5 reuses d_out (dead before decode)

  init_ws_kernel<<<1, 1, 0, stream>>>(ws);
  encode_kernel<<<B_TOT / 256, 256, 0, stream>>>(x, yst);
  for (int s = 0; s < MAX_STEPS; ++s) {
    bigstep_kernel<<<NBLK, 256, 0, stream>>>(W1, b1, W2, b2, W3, b3, ws, yst, y5);
    control_kernel<<<1, 256, 0, stream>>>(ws);
    commit_kernel<<<(B_TOT * 2) / 256, 256, 0, stream>>>(yst, y5, ws);
  }
  decode_kernel<<<B_TOT / 256, 256, 0, stream>>>(yst, Wd, bd, out);
}